// GNNModel_28509992911450
// MI455X (gfx1250) — compile-verified
//
#include <hip/hip_runtime.h>
#include <hip/hip_bf16.h>

#define N_NODES 40000
#define N_EDGES 640000
#define DF      128
#define NG      512

typedef __attribute__((ext_vector_type(2))) float v2f;
typedef __attribute__((ext_vector_type(8))) float v8f;

// ---------------------------------------------------------------------------
// Degree count: deg[col[e]] += 1
// ---------------------------------------------------------------------------
__global__ __launch_bounds__(256) void k_count_deg(const int* __restrict__ ecol,
                                                   float* __restrict__ deg) {
    int e = blockIdx.x * blockDim.x + threadIdx.x;
    if (e < N_EDGES) atomicAdd(&deg[ecol[e]], 1.0f);
}

// dinv[n] = rsqrt(1 + deg[n])
__global__ __launch_bounds__(256) void k_dinv(const float* __restrict__ deg,
                                              float* __restrict__ dinv) {
    int n = blockIdx.x * blockDim.x + threadIdx.x;
    if (n < N_NODES) dinv[n] = rsqrtf(1.0f + deg[n]);
}

// ---------------------------------------------------------------------------
// h[M,128] = A[M,128] @ W[128,128]  via V_WMMA_F32_16X16X4_F32.
// Block = 256 threads = 8 waves; block covers 128 rows (16 per wave), each
// wave computes its full 16x128 output strip.
//
// W is staged in 64KB LDS in a PAIR-INTERLEAVED layout:
//   Wl[(p*128 + n)*2 + (k&1)]  with p = k>>1
// so the B fragment for WMMA (rows k+kb, k+kb+1 of column n; k+kb always
// even) is one aligned ds_load_b64 directly into the operand register pair.
// ---------------------------------------------------------------------------
__global__ __launch_bounds__(256) void k_gemm128(const float* __restrict__ A,
                                                 const float* __restrict__ W,
                                                 float* __restrict__ out,
                                                 int M) {
    __shared__ float Wl[DF * DF]; // 64 KB, pair-interleaved

    const int t = threadIdx.x;
    // cooperative stage: read row-major, scatter into pair-interleaved layout
    for (int i0 = t * 4; i0 < DF * DF; i0 += 256 * 4) {
        float4 w = *(const float4*)&W[i0];
        int k  = i0 >> 7;         // K row
        int n0 = i0 & 127;        // first column of the 4
        float* base = &Wl[(((k >> 1) * DF) + n0) * 2 + (k & 1)];
        base[0] = w.x;
        base[2] = w.y;
        base[4] = w.z;
        base[6] = w.w;
    }
    __syncthreads();

    const int wave = t >> 5;
    const int lane = t & 31;
    const int m0   = blockIdx.x * 128 + wave * 16;

    // A-fragment addressing per ISA 32-bit 16x4 layout:
    // lanes 0-15: M=lane, K=k+0/k+1 ; lanes 16-31: M=lane-16, K=k+2/k+3
    const int ml = lane & 15;
    const int kb = (lane >> 4) << 1;

    int arow = m0 + ml;
    if (arow >= M) arow = M - 1;               // clamp (stores are guarded)
    const float* aptr = A + (size_t)arow * DF + kb;

    v8f acc[8] = {};                           // 8 N-tiles of 16

    for (int k = 0; k < DF; k += 4) {
        v2f a = *(const v2f*)(aptr + k);       // global_load_b64
        const int prow = ((k + kb) >> 1) * DF; // pair row base
#pragma unroll
        for (int nt = 0; nt < 8; ++nt) {
            // single ds_load_b64: {W[k+kb][n], W[k+kb+1][n]}
            v2f b = *(const v2f*)&Wl[(prow + nt * 16 + ml) * 2];
            acc[nt] = __builtin_amdgcn_wmma_f32_16x16x4_f32(
                false, a, false, b, (short)0, acc[nt], false, false);
        }
    }

    // C/D layout: VGPR r -> lanes 0-15: M=r, N=lane ; lanes 16-31: M=r+8
    const int rbase = (lane < 16) ? 0 : 8;
#pragma unroll
    for (int nt = 0; nt < 8; ++nt) {
#pragma unroll
        for (int r = 0; r < 8; ++r) {
            int row = m0 + rbase + r;
            if (row < M) out[(size_t)row * DF + nt * 16 + ml] = acc[nt][r];
        }
    }
}

// ---------------------------------------------------------------------------
// Self-loop init: out[n,f] = h[n,f] * dinv[n]^2  (also zero-inits scatter buf)
// ---------------------------------------------------------------------------
__global__ __launch_bounds__(256) void k_selfinit(const float* __restrict__ h,
                                                  const float* __restrict__ dinv,
                                                  float* __restrict__ out) {
    int i = blockIdx.x * blockDim.x + threadIdx.x;
    if (i >= N_NODES * DF) return;
    float d = dinv[i >> 7];
    out[i] = h[i] * d * d;
}

// ---------------------------------------------------------------------------
// Edge scatter: one wave per edge; lane handles 4 features.
// out[col] += h[row] * dinv[row]*dinv[col]
// ---------------------------------------------------------------------------
__global__ __launch_bounds__(256) void k_edge_agg(const float* __restrict__ h,
                                                  const int* __restrict__ erow,
                                                  const int* __restrict__ ecol,
                                                  const float* __restrict__ dinv,
                                                  float* __restrict__ out) {
    int e = blockIdx.x * 8 + (threadIdx.x >> 5);
    if (e >= N_EDGES) return;
    int lane = threadIdx.x & 31;
    int r = erow[e];
    int c = ecol[e];
    float norm = dinv[r] * dinv[c];
    float4 v = *(const float4*)&h[(size_t)r * DF + lane * 4];
    float* dst = &out[(size_t)c * DF + lane * 4];
    atomicAdd(dst + 0, v.x * norm);
    atomicAdd(dst + 1, v.y * norm);
    atomicAdd(dst + 2, v.z * norm);
    atomicAdd(dst + 3, v.w * norm);
}

// ---------------------------------------------------------------------------
// out[n,f] = relu(in[n,f] + b[f])
// ---------------------------------------------------------------------------
__global__ __launch_bounds__(256) void k_bias_relu(const float* __restrict__ in,
                                                   const float* __restrict__ b,
                                                   float* __restrict__ out) {
    int i = blockIdx.x * blockDim.x + threadIdx.x;
    if (i >= N_NODES * DF) return;
    out[i] = fmaxf(in[i] + b[i & 127], 0.0f);
}

// ---------------------------------------------------------------------------
// Pool: gsum[batch[n],f] += h[n,f]; gcnt[batch[n]] += 1
// ---------------------------------------------------------------------------
__global__ __launch_bounds__(256) void k_node_pool(const float* __restrict__ h,
                                                   const int* __restrict__ batch,
                                                   float* __restrict__ gsum,
                                                   float* __restrict__ gcnt) {
    int i = blockIdx.x * blockDim.x + threadIdx.x;
    if (i >= N_NODES * DF) return;
    int n = i >> 7;
    int f = i & 127;
    int g = batch[n];
    atomicAdd(&gsum[(size_t)g * DF + f], h[i]);
    if (f == 0) atomicAdd(&gcnt[g], 1.0f);
}

// ---------------------------------------------------------------------------
// Final: one wave32 per graph; out[g] = (sum_f gsum[g,f]*Wfc[f]) / max(cnt,1) + bfc
// ---------------------------------------------------------------------------
__global__ __launch_bounds__(256) void k_final(const float* __restrict__ gsum,
                                               const float* __restrict__ gcnt,
                                               const float* __restrict__ Wfc,
                                               const float* __restrict__ bfc,
                                               float* __restrict__ out) {
    int wid  = (blockIdx.x * blockDim.x + threadIdx.x) >> 5;
    int lane = threadIdx.x & 31;
    if (wid >= NG) return;
    float s = 0.0f;
#pragma unroll
    for (int i = 0; i < 4; ++i) {
        int f = lane * 4 + i;
        s += gsum[(size_t)wid * DF + f] * Wfc[f];
    }
#pragma unroll
    for (int off = 16; off > 0; off >>= 1) s += __shfl_xor(s, off, 32);
    if (lane == 0) out[wid] = s / fmaxf(gcnt[wid], 1.0f) + bfc[0];
}

// ---------------------------------------------------------------------------
extern "C" void kernel_launch(void* const* d_in, const int* in_sizes, int n_in,
                              void* d_out, int out_size, void* d_ws, size_t ws_size,
                              hipStream_t stream) {
    const float* x     = (const float*)d_in[0];
    const int*   ei    = (const int*)  d_in[1];
    const int*   batch = (const int*)  d_in[2];
    const float* W1    = (const float*)d_in[3];
    const float* b1    = (const float*)d_in[4];
    const float* W2    = (const float*)d_in[5];
    const float* b2    = (const float*)d_in[6];
    const float* Wfc   = (const float*)d_in[7];
    const float* bfc   = (const float*)d_in[8];
    float* out = (float*)d_out;

    const int* erow = ei;
    const int* ecol = ei + N_EDGES;

    // workspace layout (floats)
    float* ws   = (float*)d_ws;
    float* deg  = ws;                       // 40000
    float* dinv = ws + 40000;               // 40000
    float* bufA = ws + 80000;               // 5,120,000
    float* bufB = bufA + (size_t)N_NODES * DF;
    float* gsum = bufB + (size_t)N_NODES * DF; // 65536
    float* gcnt = gsum + (size_t)NG * DF;      // 512

    const int nf_blocks   = (N_NODES * DF + 255) / 256; // 20000
    const int gemm_blocks = (N_NODES + 127) / 128;      // 313
    const int edge_blocks = (N_EDGES + 7) / 8;          // 80000

    // degree / normalization
    hipMemsetAsync(deg, 0, N_NODES * sizeof(float), stream);
    k_count_deg<<<(N_EDGES + 255) / 256, 256, 0, stream>>>(ecol, deg);
    k_dinv<<<(N_NODES + 255) / 256, 256, 0, stream>>>(deg, dinv);

    // layer 1
    k_gemm128 <<<gemm_blocks, 256, 0, stream>>>(x, W1, bufA, N_NODES);
    k_selfinit<<<nf_blocks,   256, 0, stream>>>(bufA, dinv, bufB);
    k_edge_agg<<<edge_blocks, 256, 0, stream>>>(bufA, erow, ecol, dinv, bufB);
    k_bias_relu<<<nf_blocks,  256, 0, stream>>>(bufB, b1, bufA);

    // layer 2
    k_gemm128 <<<gemm_blocks, 256, 0, stream>>>(bufA, W2, bufB, N_NODES);
    k_selfinit<<<nf_blocks,   256, 0, stream>>>(bufB, dinv, bufA);
    k_edge_agg<<<edge_blocks, 256, 0, stream>>>(bufB, erow, ecol, dinv, bufA);
    k_bias_relu<<<nf_blocks,  256, 0, stream>>>(bufA, b2, bufB);

    // mean pool + fc
    hipMemsetAsync(gsum, 0, (NG * DF + NG) * sizeof(float), stream);
    k_node_pool<<<nf_blocks, 256, 0, stream>>>(bufB, batch, gsum, gcnt);
    k_final<<<(NG * 32 + 255) / 256, 256, 0, stream>>>(gsum, gcnt, Wfc, bfc, out);
}